// TorchCombineModule_64201171140968
// MI455X (gfx1250) — compile-verified
//
#include <hip/hip_runtime.h>
#include <hip/hip_bf16.h>
#include <stdint.h>

// Problem constants from the reference module.
#define G_    1
#define C_    8
#define EPC_  8
#define K_    8
#define S_    4096
#define NR_   (G_ * C_ * EPC_)   // 64
#define P_    36864
#define EMB_  1024
#define ROW_BYTES (EMB_ * 2)     // 2048 bytes per bf16 row

#define WAVES_PER_BLOCK 8
#define THREADS (WAVES_PER_BLOCK * 32)

typedef uint32_t v4u __attribute__((ext_vector_type(4)));
typedef int      v4i __attribute__((ext_vector_type(4)));

#if defined(__gfx1250__) && \
    __has_builtin(__builtin_amdgcn_global_load_async_to_lds_b128) && \
    __has_builtin(__builtin_amdgcn_global_store_async_from_lds_b128)
#define USE_ASYNC_LDS 1
typedef __attribute__((address_space(1))) v4i gv4i;   // global int4 (AS1)
typedef __attribute__((address_space(3))) v4i lv4i;   // LDS int4 (AS3)
// CPol: TH[2:0]=1 (non-temporal) — both streams exceed the 192 MB L2 and are
// touched exactly once, so bypass temporal caching.
#define CPOL_NT 1
#else
#define USE_ASYNC_LDS 0
#endif

__global__ __launch_bounds__(THREADS)
void combine_scatter(const __hip_bfloat16* __restrict__ buf,
                     const int* __restrict__ metadata,
                     const int* __restrict__ counts,
                     const int* __restrict__ offsets,
                     __hip_bfloat16* __restrict__ out)
{
#if USE_ASYNC_LDS
    __shared__ __align__(16) uint8_t smem[WAVES_PER_BLOCK * ROW_BYTES];
#endif
    const int lane = threadIdx.x & 31;
    // Wave index is uniform within the wave: pin it to an SGPR so the whole
    // validity / metadata / destination-index path scalarizes (s_load + s_cmp
    // + s_cbranch instead of v_cmp chains and vector 64-bit address math).
    const int wave = __builtin_amdgcn_readfirstlane(threadIdx.x >> 5);
    const int c    = blockIdx.y;                          // chip
    const int p    = blockIdx.x * WAVES_PER_BLOCK + wave; // buffer slot (scalar)

    // Validity: p inside one of chip c's 8 expert regions (scalar loads).
    const int* off_c = offsets + c * NR_ + c * EPC_;
    const int* cnt_c = counts + c * EPC_;
    bool valid = false;
#pragma unroll
    for (int e = 0; e < EPC_; ++e) {
        const int s = off_c[e];
        const int n = cnt_c[e];
        valid |= (p >= s) & (p < s + n);
    }
    if (!valid) return;  // scalar, wave-uniform exit

    const int* m = metadata + ((size_t)c * P_ + p) * 3;
    const int src_chip = m[0];   // G==1: meta0 % G == 0 always; src_chip = meta0
    const int token    = m[1];
    const int topk     = m[2];

    const size_t dst_row = (size_t)src_chip * (S_ * K_) + (size_t)token * K_ + (size_t)topk;
    const uint8_t* __restrict__ src = (const uint8_t*)(buf + ((size_t)c * P_ + p) * EMB_);
    uint8_t* __restrict__       dst = (uint8_t*)(out + dst_row * EMB_);

#if USE_ASYNC_LDS
    // Memory -> LDS -> memory via gfx1250 async DMA ops; no VGPR data round
    // trip. 32 lanes x 4 x B128 = 2048 B per row; ASYNCcnt orders the phases.
    gv4i* gsrc = (gv4i*)src;
    gv4i* gdst = (gv4i*)dst;
    lv4i* lrow = (lv4i*)&smem[wave * ROW_BYTES];
#pragma unroll
    for (int j = 0; j < 4; ++j) {
        const int o = lane + j * 32;   // 16-byte vector index
        __builtin_amdgcn_global_load_async_to_lds_b128(gsrc + o, lrow + o, 0, CPOL_NT);
    }
    // Data must land in LDS before the async stores read it (LDS side of
    // async ops is not ordered between instructions).
#if __has_builtin(__builtin_amdgcn_s_wait_asynccnt)
    __builtin_amdgcn_s_wait_asynccnt(0);
#else
    asm volatile("s_wait_asynccnt 0" ::: "memory");
#endif
#pragma unroll
    for (int j = 0; j < 4; ++j) {
        const int o = lane + j * 32;
        __builtin_amdgcn_global_store_async_from_lds_b128(gdst + o, lrow + o, 0, CPOL_NT);
    }
    // S_ENDPGM performs an implicit wait-idle; LDS slice is private per wave.
#else
    // Direct path: 4 x 128-bit nontemporal load/store per lane.
    const v4u* __restrict__ s4 = (const v4u*)src;
    v4u* __restrict__       d4 = (v4u*)dst;
    v4u r0 = __builtin_nontemporal_load(&s4[lane + 0 * 32]);
    v4u r1 = __builtin_nontemporal_load(&s4[lane + 1 * 32]);
    v4u r2 = __builtin_nontemporal_load(&s4[lane + 2 * 32]);
    v4u r3 = __builtin_nontemporal_load(&s4[lane + 3 * 32]);
    __builtin_nontemporal_store(r0, &d4[lane + 0 * 32]);
    __builtin_nontemporal_store(r1, &d4[lane + 1 * 32]);
    __builtin_nontemporal_store(r2, &d4[lane + 2 * 32]);
    __builtin_nontemporal_store(r3, &d4[lane + 3 * 32]);
#endif
}

extern "C" void kernel_launch(void* const* d_in, const int* in_sizes, int n_in,
                              void* d_out, int out_size, void* d_ws, size_t ws_size,
                              hipStream_t stream)
{
    (void)in_sizes; (void)n_in; (void)out_size; (void)d_ws; (void)ws_size;
    const __hip_bfloat16* buf      = (const __hip_bfloat16*)d_in[0];
    const int*            metadata = (const int*)d_in[1];
    const int*            counts   = (const int*)d_in[2];
    const int*            offsets  = (const int*)d_in[3];
    __hip_bfloat16*       out      = (__hip_bfloat16*)d_out;

    // Every output row receives exactly one write (counts sum to C*S*K and all
    // (src_chip, token, topk) triples are distinct), so no zero-fill pass.
    dim3 grid(P_ / WAVES_PER_BLOCK, C_, 1);
    combine_scatter<<<grid, THREADS, 0, stream>>>(buf, metadata, counts, offsets, out);
}